// UnconditionalPRADA_369367188159
// MI455X (gfx1250) — compile-verified
//
#include <hip/hip_runtime.h>
#include <hip/hip_bf16.h>

typedef float v2f __attribute__((ext_vector_type(2)));
typedef float v8f __attribute__((ext_vector_type(8)));

#define FDIM 128
#define HDIM 128
#define LDIM 64

static inline int cdiv(long a, long b) { return (int)((a + b - 1) / b); }

// ---------------- degree / normalization ----------------
__global__ void k_fill1(float* __restrict__ deg, int n) {
    int i = blockIdx.x * blockDim.x + threadIdx.x;
    if (i < n) deg[i] = 1.0f;  // self loop
}

__global__ void k_degcount(const int* __restrict__ dst, float* __restrict__ deg, int e) {
    int i = blockIdx.x * blockDim.x + threadIdx.x;
    if (i < e) atomicAdd(&deg[dst[i]], 1.0f);
}

__global__ void k_rsqrt_inplace(float* __restrict__ d, int n) {
    int i = blockIdx.x * blockDim.x + threadIdx.x;
    if (i < n) d[i] = rsqrtf(d[i]);
}

__global__ void k_edgenorm(const int* __restrict__ src, const int* __restrict__ dst,
                           const float* __restrict__ dinv, float* __restrict__ norm, int e) {
    int i = blockIdx.x * blockDim.x + threadIdx.x;
    if (i < e) norm[i] = dinv[src[i]] * dinv[dst[i]];
}

// ---------------- WMMA f32 GEMM: Y[n,O] = X[n,K] @ W[K,O] ----------------
// One wave computes a 16-row tile across all O columns using V_WMMA_F32_16X16X4_F32.
// A (16x4 f32):  lane m=l&15 holds row m; VGPR0 = K {0 | 2} by half-wave, VGPR1 = K {1 | 3}.
// B (4x16 f32):  lane n=l&15 holds col n; same half-wave K split.
// D (16x16 f32): VGPR r -> row r (lanes 0-15) / row r+8 (lanes 16-31), col = l&15.
template <int K, int O>
__global__ void __launch_bounds__(128)
k_gemm_wmma(const float* __restrict__ X, const float* __restrict__ W,
            float* __restrict__ Y, int n) {
    const int lane = threadIdx.x & 31;
    const int wave = threadIdx.x >> 5;
    const int tile = blockIdx.x * 4 + wave;
    const int nTiles = (n + 15) >> 4;
    if (tile >= nTiles) return;           // wave-uniform: EXEC stays all-ones for WMMA

    const int row0 = tile << 4;
    const int m    = lane & 15;
    const int kb   = (lane >> 4) << 1;    // 0 or 2 (half-wave K split)
    const int col  = lane & 15;

    constexpr int NT = O / 16;
    v8f acc[NT];
#pragma unroll
    for (int t = 0; t < NT; ++t) acc[t] = (v8f){0.f,0.f,0.f,0.f,0.f,0.f,0.f,0.f};

    int rowA = row0 + m;
    if (rowA >= n) rowA = n - 1;          // clamp (duplicate read is harmless)
    const float* __restrict__ xrow = X + (long)rowA * K;

#pragma unroll 2
    for (int k0 = 0; k0 < K; k0 += 4) {
        v2f a;
        a.x = xrow[k0 + kb];
        a.y = xrow[k0 + kb + 1];
        const float* __restrict__ w0 = W + (long)(k0 + kb) * O;
        const float* __restrict__ w1 = w0 + O;
#pragma unroll
        for (int t = 0; t < NT; ++t) {
            v2f b;
            b.x = w0[t * 16 + col];
            b.y = w1[t * 16 + col];
            acc[t] = __builtin_amdgcn_wmma_f32_16x16x4_f32(
                false, a, false, b, (short)0, acc[t], false, false);
        }
    }

    const int mbase = (lane >> 4) << 3;   // 0 or 8
#pragma unroll
    for (int t = 0; t < NT; ++t) {
#pragma unroll
        for (int r = 0; r < 8; ++r) {
            int row = row0 + mbase + r;
            if (row < n) Y[(long)row * O + t * 16 + col] = acc[t][r];
        }
    }
}

// ---------------- aggregation ----------------
// out[i,f] = xw[i,f] * dinv[i]^2 + bias[f]   (self-loop + bias; fully initializes out)
template <int DIM>
__global__ void k_selfinit(const float* __restrict__ xw, const float* __restrict__ dinv,
                           const float* __restrict__ bias, float* __restrict__ out, int n) {
    long i = (long)blockIdx.x * blockDim.x + threadIdx.x;
    if (i >= (long)n * DIM) return;
    int node = (int)(i / DIM);
    int f    = (int)(i % DIM);
    float d = dinv[node];
    out[i] = xw[i] * d * d + bias[f];
}

// out[dst] += xw[src] * norm[e]  (float4 chunks, fp32 L2 atomics)
template <int DIM>
__global__ void k_scatter(const float* __restrict__ xw, const float* __restrict__ norm,
                          const int* __restrict__ src, const int* __restrict__ dst,
                          float* __restrict__ out, int e) {
    constexpr int C = DIM / 4;
    long gid = (long)blockIdx.x * blockDim.x + threadIdx.x;
    if (gid >= (long)e * C) return;
    int edge = (int)(gid / C);
    int c    = (int)(gid % C);
    int s = src[edge];
    int d = dst[edge];
    float nm = norm[edge];
    float4 v = ((const float4*)(xw + (long)s * DIM))[c];
    float* o = out + (long)d * DIM + c * 4;
    atomicAdd(o + 0, v.x * nm);
    atomicAdd(o + 1, v.y * nm);
    atomicAdd(o + 2, v.z * nm);
    atomicAdd(o + 3, v.w * nm);
}

// ---------------- elementwise ----------------
__global__ void k_tanh(float* __restrict__ x, long n) {
    long i = (long)blockIdx.x * blockDim.x + threadIdx.x;
    if (i < n) x[i] = tanhf(x[i]);
}

__global__ void k_reparam(const float* __restrict__ noise, const float* __restrict__ logv,
                          const float* __restrict__ mean, float* __restrict__ z, long n) {
    long i = (long)blockIdx.x * blockDim.x + threadIdx.x;
    if (i < n) z[i] = noise[i] * __expf(0.5f * logv[i]) + mean[i];
}

// ---------------- driver ----------------
extern "C" void kernel_launch(void* const* d_in, const int* in_sizes, int n_in,
                              void* d_out, int out_size, void* d_ws, size_t ws_size,
                              hipStream_t stream) {
    const float* feature = (const float*)d_in[0];
    const int*   edge    = (const int*)d_in[1];
    const float* noise   = (const float*)d_in[2];
    const float* W_enc   = (const float*)d_in[3];
    const float* b_enc   = (const float*)d_in[4];
    const float* W_mu    = (const float*)d_in[5];
    const float* b_mu    = (const float*)d_in[6];
    const float* W_lv    = (const float*)d_in[7];
    const float* b_lv    = (const float*)d_in[8];
    const float* W_d1    = (const float*)d_in[9];
    const float* b_d1    = (const float*)d_in[10];
    const float* W_d2    = (const float*)d_in[11];
    const float* b_d2    = (const float*)d_in[12];

    const int N = in_sizes[0] / FDIM;
    const int E = in_sizes[1] / 2;
    const int* src = edge;
    const int* dst = edge + E;

    // workspace layout
    float* ws   = (float*)d_ws;
    float* dinv = ws;                        // N
    float* enorm = dinv + N;                 // E
    float* bufA = enorm + E;                 // N*128  (gemm output)
    float* bufB = bufA + (long)N * 128;      // N*128  (h / hd)
    float* bufC = bufB + (long)N * 128;      // N*64   (xw for mu)
    float* bufD = bufC + (long)N * 64;       // N*64   (xw for lv)

    // output layout: z | mean | logvar | out
    float* z    = (float*)d_out;
    float* mean = z    + (long)N * LDIM;
    float* logv = mean + (long)N * LDIM;
    float* outF = logv + (long)N * LDIM;

    const int B = 256;
    const int nTiles = (N + 15) / 16;
    const dim3 gGemm(cdiv(nTiles, 4));

    // degrees + per-edge norm (shared by all 5 convs)
    k_fill1<<<cdiv(N, B), B, 0, stream>>>(dinv, N);
    k_degcount<<<cdiv(E, B), B, 0, stream>>>(dst, dinv, E);
    k_rsqrt_inplace<<<cdiv(N, B), B, 0, stream>>>(dinv, N);
    k_edgenorm<<<cdiv(E, B), B, 0, stream>>>(src, dst, dinv, enorm, E);

    // ---- encoder conv: h = tanh(gcn(feature, W_enc, b_enc)) ----
    k_gemm_wmma<FDIM, HDIM><<<gGemm, 128, 0, stream>>>(feature, W_enc, bufA, N);
    k_selfinit<HDIM><<<cdiv((long)N * HDIM, B), B, 0, stream>>>(bufA, dinv, b_enc, bufB, N);
    k_scatter<HDIM><<<cdiv((long)E * (HDIM / 4), B), B, 0, stream>>>(bufA, enorm, src, dst, bufB, E);
    k_tanh<<<cdiv((long)N * HDIM, B), B, 0, stream>>>(bufB, (long)N * HDIM);

    // ---- mean = gcn(h, W_mu, b_mu) ----
    k_gemm_wmma<HDIM, LDIM><<<gGemm, 128, 0, stream>>>(bufB, W_mu, bufC, N);
    k_selfinit<LDIM><<<cdiv((long)N * LDIM, B), B, 0, stream>>>(bufC, dinv, b_mu, mean, N);
    k_scatter<LDIM><<<cdiv((long)E * (LDIM / 4), B), B, 0, stream>>>(bufC, enorm, src, dst, mean, E);

    // ---- logvar = gcn(h, W_lv, b_lv) ----
    k_gemm_wmma<HDIM, LDIM><<<gGemm, 128, 0, stream>>>(bufB, W_lv, bufD, N);
    k_selfinit<LDIM><<<cdiv((long)N * LDIM, B), B, 0, stream>>>(bufD, dinv, b_lv, logv, N);
    k_scatter<LDIM><<<cdiv((long)E * (LDIM / 4), B), B, 0, stream>>>(bufD, enorm, src, dst, logv, E);

    // ---- z = noise * exp(0.5*logvar) + mean ----
    k_reparam<<<cdiv((long)N * LDIM, B), B, 0, stream>>>(noise, logv, mean, z, (long)N * LDIM);

    // ---- decoder conv 1: hd = tanh(gcn(z, W_d1, b_d1)) ----
    k_gemm_wmma<LDIM, HDIM><<<gGemm, 128, 0, stream>>>(z, W_d1, bufA, N);
    k_selfinit<HDIM><<<cdiv((long)N * HDIM, B), B, 0, stream>>>(bufA, dinv, b_d1, bufB, N);
    k_scatter<HDIM><<<cdiv((long)E * (HDIM / 4), B), B, 0, stream>>>(bufA, enorm, src, dst, bufB, E);
    k_tanh<<<cdiv((long)N * HDIM, B), B, 0, stream>>>(bufB, (long)N * HDIM);

    // ---- decoder conv 2: out = gcn(hd, W_d2, b_d2) ----
    k_gemm_wmma<HDIM, FDIM><<<gGemm, 128, 0, stream>>>(bufB, W_d2, bufA, N);
    k_selfinit<FDIM><<<cdiv((long)N * FDIM, B), B, 0, stream>>>(bufA, dinv, b_d2, outF, N);
    k_scatter<FDIM><<<cdiv((long)E * (FDIM / 4), B), B, 0, stream>>>(bufA, enorm, src, dst, outF, E);
}